// CapsuleLayer_16449724744202
// MI455X (gfx1250) — compile-verified
//
#include <hip/hip_runtime.h>
#include <math.h>

// ---------------------------------------------------------------------------
// CapsuleLayer dynamic routing for MI455X (gfx1250, wave32).
//
// Plan:
//  K1: priors[n,b,r,o] via V_WMMA_F32_16X16X4_F32 (M=16 b's, N=16 o's, K=8 as
//      two chained K=4 WMMAs). fp32 matrix pipe => bit-compatible FMA math.
//  K2: one 256-thread workgroup (8 wave32s) per (n,b); copy the 73.7KB
//      priors slab into LDS once (async load-to-LDS when available), run all
//      3 routing rounds from LDS.
//  Fallback (tiny workspace): fused VALU kernel, priors built in LDS.
// ---------------------------------------------------------------------------

typedef __attribute__((ext_vector_type(2))) float v2f;
typedef __attribute__((ext_vector_type(8))) float v8f;
typedef __attribute__((__vector_size__(4 * sizeof(int)))) int v4i;

#define CAPS_N   10
#define CAPS_B   512
#define CAPS_R   1152
#define CAPS_CI  8
#define CAPS_CO  16

#define CAPS_GLOBAL_AS __attribute__((address_space(1)))
#define CAPS_LDS_AS    __attribute__((address_space(3)))

#if defined(__has_builtin)
#if __has_builtin(__builtin_amdgcn_global_load_async_to_lds_b128)
#define CAPS_ASYNC_LDS 1
#endif
#endif

// ------------------------- Kernel 1: priors via WMMA ------------------------
// grid.x = CAPS_N * nBtiles, block = 256 (8 waves). Each wave owns r = wave+8k.
// Per (n, r, b-tile16): D[16,16] = x[btile,r,0:8] (16x8) @ W[n,r] (8x16).
__global__ __launch_bounds__(256) void CapsPriors_wmma_kernel(
    const float* __restrict__ x, const float* __restrict__ W,
    float* __restrict__ priors, int Bc, int bBase, int nBtiles)
{
    const int R = CAPS_R;
    const int n    = blockIdx.x / nBtiles;
    const int bt   = blockIdx.x % nBtiles;
    const int t    = threadIdx.x;
    const int lane = t & 31;
    const int wave = t >> 5;
    const int m16  = lane & 15;   // A-row (b) and B/D-col (o)
    const int hi   = lane >> 4;   // K half selector

    // A: x[bA, r, i], lane holds K = {2hi, 2hi+1} (wmma1) and +4 (wmma2)
    const int bA = bBase + bt * 16 + m16;
    const float* xbase = x + (size_t)bA * R * 8 + 2 * hi;
    // B: W[n, r, i, o], i = {2hi, 2hi+1} (+4), o = m16
    const float* wbase = W + ((size_t)n * R * 8 + 2 * hi) * 16 + m16;
    // D: VGPR j, lane -> priors[bLocal = bt*16 + 8*hi + j][o = m16]
    const int bD = bt * 16 + hi * 8;
    float* pbase = priors + ((size_t)(n * Bc + bD) * R) * 16 + m16;

    for (int r = wave; r < R; r += 8) {
        const float* xp = xbase + (size_t)r * 8;
        v2f a0 = *(const v2f*)(xp);        // K = 2hi, 2hi+1
        v2f a1 = *(const v2f*)(xp + 4);    // K = 2hi+4, 2hi+5
        const float* wp = wbase + (size_t)r * 128;
        v2f b0 = { wp[0],  wp[16] };       // rows i = 2hi, 2hi+1
        v2f b1 = { wp[64], wp[80] };       // rows i = 2hi+4, 2hi+5
        v8f c = {};
        c = __builtin_amdgcn_wmma_f32_16x16x4_f32(false, a0, false, b0,
                                                  (short)0, c, false, false);
        c = __builtin_amdgcn_wmma_f32_16x16x4_f32(false, a1, false, b1,
                                                  (short)0, c, false, false);
        float* pp = pbase + (size_t)r * 16;
#pragma unroll
        for (int j = 0; j < 8; ++j)
            pp[(size_t)j * (CAPS_R * 16)] = c[j];   // b-stride = R*16 floats
    }
}

// ------------------------- Routing body (shared) ----------------------------
// P[R*16] priors slab in LDS, 256 threads, 3 routing rounds, writes 16 outputs.
__device__ __forceinline__ void routing_body(
    float* __restrict__ P, float* __restrict__ logitsArr,
    float* __restrict__ evals, float* __restrict__ red,
    float* __restrict__ svec, float* __restrict__ ovec,
    float* __restrict__ outp, int t)
{
    const int R = CAPS_R;
    const int o = t & 15;
    const int g = t >> 4;

    for (int round = 0; round < 3; ++round) {
        float acc = 0.0f;
        float scale;
        if (round == 0) {
            // softmax(0) over R is uniform: probs = 1/R
            for (int k = 0; k < R / 16; ++k)
                acc += P[(g + 16 * k) * 16 + o];
            scale = 1.0f / (float)R;
        } else {
            // softmax over R of logits (uniform over Cout in the reference)
            float m = -1e30f;
            for (int r = t; r < R; r += 256) m = fmaxf(m, logitsArr[r]);
            red[t] = m; __syncthreads();
            for (int s = 128; s > 0; s >>= 1) {
                if (t < s) red[t] = fmaxf(red[t], red[t + s]);
                __syncthreads();
            }
            const float mx = red[0]; __syncthreads();
            float se = 0.0f;
            for (int r = t; r < R; r += 256) {
                float e = expf(logitsArr[r] - mx);
                evals[r] = e;
                se += e;
            }
            red[t] = se; __syncthreads();
            for (int s = 128; s > 0; s >>= 1) {
                if (t < s) red[t] += red[t + s];
                __syncthreads();
            }
            const float sumE = red[0]; __syncthreads();
            for (int k = 0; k < R / 16; ++k) {
                int r = g + 16 * k;
                acc += evals[r] * P[r * 16 + o];
            }
            scale = 1.0f / sumE;
        }
        // reduce 16 partials per o, then squash
        red[t] = acc; __syncthreads();
        if (t < 16) {
            float s = 0.0f;
            for (int gg = 0; gg < 16; ++gg) s += red[gg * 16 + t];
            svec[t] = s * scale;
        }
        __syncthreads();
        float s_norm = 0.0f;
        for (int oo = 0; oo < 16; ++oo) { float v = svec[oo]; s_norm += v * v; }
        const float factor = (s_norm / (1.0f + s_norm)) / sqrtf(s_norm);
        if (t < 16) ovec[t] = svec[t] * factor;
        __syncthreads();
        if (round < 2) {
            // logits[r] += dot(priors[r,:], outp)
            for (int r = t; r < R; r += 256) {
                float d = 0.0f;
                for (int oo = 0; oo < 16; ++oo) d += P[r * 16 + oo] * ovec[oo];
                logitsArr[r] = (round == 0) ? d : (logitsArr[r] + d);
            }
            __syncthreads();
        }
    }
    if (t < 16) outp[t] = ovec[t];
}

// ------------------------- Kernel 2: routing from workspace -----------------
__global__ __launch_bounds__(256, 1) void CapsRouting_kernel(
    const float* __restrict__ priors, float* __restrict__ out,
    int Bc, int bBase)
{
    __shared__ __align__(16) float P[CAPS_R * 16];   // 73728 B
    __shared__ float logitsArr[CAPS_R];              //  4608 B
    __shared__ float evals[CAPS_R];                  //  4608 B
    __shared__ float red[256];
    __shared__ float svec[16];
    __shared__ float ovec[16];

    const int n  = blockIdx.x / Bc;
    const int bL = blockIdx.x % Bc;
    const int t  = threadIdx.x;

    const float* src = priors + (size_t)(n * Bc + bL) * CAPS_R * 16;

#if defined(CAPS_ASYNC_LDS)
    {
        // CDNA5 async load-to-LDS: bypass VGPR staging, tracked by ASYNCcnt.
        const CAPS_GLOBAL_AS char* g =
            (const CAPS_GLOBAL_AS char*)(const void*)src;
        CAPS_LDS_AS char* l = (CAPS_LDS_AS char*)(void*)P;
        for (int i = t; i < CAPS_R * 16 / 4; i += 256)
            __builtin_amdgcn_global_load_async_to_lds_b128(
                (CAPS_GLOBAL_AS v4i*)(g + (size_t)i * 16),
                (CAPS_LDS_AS v4i*)(l + (size_t)i * 16),
                /*imm offset=*/0, /*cpol=*/0);
#if defined(__has_builtin) && __has_builtin(__builtin_amdgcn_s_wait_asynccnt)
        __builtin_amdgcn_s_wait_asynccnt(0);
#else
        asm volatile("s_wait_asynccnt 0" ::: "memory");
#endif
    }
    __syncthreads();
#else
    for (int i = t; i < CAPS_R * 16 / 4; i += 256)
        ((float4*)P)[i] = ((const float4*)src)[i];
    __syncthreads();
#endif

    routing_body(P, logitsArr, evals, red, svec, ovec,
                 out + ((size_t)n * CAPS_B + bBase + bL) * 16, t);
}

// ------------------------- Fallback: fused, no workspace --------------------
__global__ __launch_bounds__(256, 1) void CapsFused_kernel(
    const float* __restrict__ x, const float* __restrict__ W,
    float* __restrict__ out)
{
    __shared__ __align__(16) float P[CAPS_R * 16];   // 73728 B
    __shared__ __align__(16) float xl[CAPS_R * 8];   // 36864 B
    __shared__ float logitsArr[CAPS_R];
    __shared__ float evals[CAPS_R];
    __shared__ float red[256];
    __shared__ float svec[16];
    __shared__ float ovec[16];

    const int n = blockIdx.x >> 9;        // / 512
    const int b = blockIdx.x & 511;
    const int t = threadIdx.x;

    const float4* xsrc = (const float4*)(x + (size_t)b * CAPS_R * 8);
    for (int i = t; i < CAPS_R * 8 / 4; i += 256)
        ((float4*)xl)[i] = xsrc[i];
    __syncthreads();

    for (int idx = t; idx < CAPS_R * 16; idx += 256) {
        const int r = idx >> 4, o = idx & 15;
        const float* wr = W + ((size_t)(n * CAPS_R + r) * 8) * 16 + o;
        float s = 0.0f;
#pragma unroll
        for (int i = 0; i < 8; ++i) s += xl[r * 8 + i] * wr[i * 16];
        P[idx] = s;
    }
    __syncthreads();

    routing_body(P, logitsArr, evals, red, svec, ovec,
                 out + ((size_t)n * CAPS_B + b) * 16, t);
}

// ------------------------- Host launcher ------------------------------------
extern "C" void kernel_launch(void* const* d_in, const int* in_sizes, int n_in,
                              void* d_out, int out_size, void* d_ws, size_t ws_size,
                              hipStream_t stream)
{
    (void)in_sizes; (void)n_in; (void)out_size;
    const float* x = (const float*)d_in[0];   // [512,1152,8]
    const float* W = (const float*)d_in[1];   // [10,1152,8,16]
    float* out = (float*)d_out;               // [10,512,16]
    float* ws  = (float*)d_ws;

    // bytes of priors per single b across all n
    const size_t perB = (size_t)CAPS_N * CAPS_R * 16 * sizeof(float); // 737280
    int Bc = 0;
    for (int c = CAPS_B; c >= 16; c >>= 1) {
        if ((size_t)c * perB <= ws_size) { Bc = c; break; }
    }

    if (Bc > 0) {
        const int nBtiles = Bc / 16;
        for (int bBase = 0; bBase < CAPS_B; bBase += Bc) {
            CapsPriors_wmma_kernel<<<dim3(CAPS_N * nBtiles), dim3(256), 0, stream>>>(
                x, W, ws, Bc, bBase, nBtiles);
            CapsRouting_kernel<<<dim3(CAPS_N * Bc), dim3(256), 0, stream>>>(
                ws, out, Bc, bBase);
        }
    } else {
        CapsFused_kernel<<<dim3(CAPS_N * CAPS_B), dim3(256), 0, stream>>>(x, W, out);
    }
}